// PointerNet_73126113182238
// MI455X (gfx1250) — compile-verified
//
#include <hip/hip_runtime.h>
#include <math.h>

typedef __attribute__((ext_vector_type(2))) float v2f;
typedef __attribute__((ext_vector_type(8))) float v8f;

// Problem dims (from reference setup_inputs)
constexpr int B = 4, T = 512, I = 512, D = 512, H = 8, V = 32000;
constexpr int TI = T * I;                       // 262144
constexpr long long BTV = (long long)B * T * V; // 65,536,000

// ---------------------------------------------------------------------------
// 1) attn = mean over heads: (B,H,T,I) -> (B,T,I), float4 vectorized
// ---------------------------------------------------------------------------
__global__ void mean_heads_kernel(const float* __restrict__ heads,
                                  float* __restrict__ attn) {
    int n = blockIdx.x * blockDim.x + threadIdx.x;   // over B*T*I/4
    int b   = n / (TI / 4);
    int rem = n % (TI / 4);
    const float4* base = (const float4*)heads + (size_t)b * H * (TI / 4) + rem;
    float4 s = make_float4(0.f, 0.f, 0.f, 0.f);
#pragma unroll
    for (int h = 0; h < H; ++h) {
        float4 v = base[(size_t)h * (TI / 4)];
        s.x += v.x; s.y += v.y; s.z += v.z; s.w += v.w;
    }
    const float inv = 1.0f / (float)H;
    ((float4*)attn)[n] = make_float4(s.x * inv, s.y * inv, s.z * inv, s.w * inv);
}

// ---------------------------------------------------------------------------
// 2) context = attn @ enc  (per-batch 512x512x512 f32 GEMM)
//    One wave per 16x16 C tile; K stepped by 4 via V_WMMA_F32_16X16X4_F32.
//    A 16x4 layout: lane(hi,m): VGPR j = A[m][kb + 2*hi + j]
//    B 4x16 layout: lane(hi,n): VGPR j = B[kb + 2*hi + j][n]
//    C 16x16 layout: lane(hi,n): VGPR r = C[r + 8*hi][n]
// ---------------------------------------------------------------------------
__global__ void __launch_bounds__(128)
context_wmma_kernel(const float* __restrict__ attn,
                    const float* __restrict__ enc,
                    float* __restrict__ ctx) {
    const int lane = threadIdx.x & 31;
    const int wave = threadIdx.x >> 5;
    const int tile = blockIdx.x * 4 + wave;   // 0..4095
    const int b  = tile >> 10;                // batch
    const int tt = (tile >> 5) & 31;          // T tile
    const int dt = tile & 31;                 // D tile
    const int m  = lane & 15;                 // row (A) / col (B,C)
    const int hi = lane >> 4;

    const float* Arow  = attn + (size_t)b * TI + (size_t)(tt * 16 + m) * I + 2 * hi;
    const float* Bcol  = enc  + (size_t)b * I * D + dt * 16 + m;

    v8f c = {0.f, 0.f, 0.f, 0.f, 0.f, 0.f, 0.f, 0.f};
#pragma unroll 4
    for (int kb = 0; kb < I; kb += 4) {
        v2f a = *(const v2f*)(Arow + kb);                 // global_load_b64
        v2f bv;
        bv.x = Bcol[(size_t)(kb + 2 * hi + 0) * D];
        bv.y = Bcol[(size_t)(kb + 2 * hi + 1) * D];
        c = __builtin_amdgcn_wmma_f32_16x16x4_f32(
                /*neg_a=*/false, a, /*neg_b=*/false, bv,
                /*c_mod=*/(short)0, c, /*reuse_a=*/false, /*reuse_b=*/false);
    }

    float* Cout = ctx + (size_t)b * T * D + (size_t)(tt * 16) * D + dt * 16 + m;
#pragma unroll
    for (int r = 0; r < 8; ++r)
        Cout[(size_t)(r + 8 * hi) * D] = c[r];
}

// ---------------------------------------------------------------------------
// 3) p_gen = sigmoid([context|dec|tar] . W + b) ; one wave per (b,t)
// ---------------------------------------------------------------------------
__global__ void pgen_kernel(const float* __restrict__ ctx,
                            const float* __restrict__ dec,
                            const float* __restrict__ tar,
                            const float* __restrict__ W,
                            const float* __restrict__ bias,
                            float* __restrict__ pgen_out) {
    int wave = (blockIdx.x * blockDim.x + threadIdx.x) >> 5;  // 0..B*T-1
    int lane = threadIdx.x & 31;
    const float* c = ctx + (size_t)wave * D;
    const float* d = dec + (size_t)wave * D;
    const float* t = tar + (size_t)wave * D;
    float s = 0.f;
#pragma unroll 4
    for (int j = lane; j < D; j += 32)
        s += c[j] * W[j] + d[j] * W[D + j] + t[j] * W[2 * D + j];
#pragma unroll
    for (int off = 16; off > 0; off >>= 1)
        s += __shfl_xor(s, off, 32);
    if (lane == 0) {
        float x = s + bias[0];
        pgen_out[wave] = 1.0f / (1.0f + expf(-x));
    }
}

// ---------------------------------------------------------------------------
// 4) Fused pointer-row kernel: one workgroup per (b,t).
//    Build the 32000-float vocab row in LDS (zero + ds_add_f32 scatter of the
//    512 attention weights), then stream once over V:
//        ptr[b,t,:] = row ; fin[b,t,:] = p*gen + (1-p)*row
//    Removes the global zero pass, global atomics, and the ptr re-read:
//    786 MB total traffic instead of ~1.1 GB.
//    LDS = 128,000 B per workgroup (CDNA5: 320 KB/WGP -> 2 WGs resident).
// ---------------------------------------------------------------------------
__global__ void __launch_bounds__(512)
pointer_fuse_kernel(const float* __restrict__ attn,
                    const int* __restrict__ tokens,
                    const float* __restrict__ gen,
                    const float* __restrict__ pgen,
                    float* __restrict__ ptr,
                    float* __restrict__ fin) {
    extern __shared__ float row[];        // V floats = 128,000 bytes
    const int bt  = blockIdx.x;           // 0..B*T-1
    const int b   = bt / T;
    const int tid = threadIdx.x;          // 0..511

    // zero the LDS vocab row (ds_store_b128)
    float4* row4 = (float4*)row;
    for (int i = tid; i < V / 4; i += 512)
        row4[i] = make_float4(0.f, 0.f, 0.f, 0.f);
    __syncthreads();

    // scatter: thread tid handles source position i = tid (I == blockDim == 512)
    {
        int   tok = tokens[b * I + tid];
        float a   = attn[(size_t)bt * I + tid];
        atomicAdd(&row[tok], a);          // ds_add_f32, duplicates handled
    }
    __syncthreads();

    const float p = pgen[bt];
    const float q = 1.0f - p;
    const float4* g4 = (const float4*)(gen + (size_t)bt * V);
    float4*       p4 = (float4*)(ptr + (size_t)bt * V);
    float4*       f4 = (float4*)(fin + (size_t)bt * V);
    for (int i = tid; i < V / 4; i += 512) {
        float4 r = row4[i];
        float4 g = g4[i];
        p4[i] = r;
        float4 o;
        o.x = p * g.x + q * r.x;
        o.y = p * g.y + q * r.y;
        o.z = p * g.z + q * r.z;
        o.w = p * g.w + q * r.w;
        f4[i] = o;
    }
}

// ---------------------------------------------------------------------------
extern "C" void kernel_launch(void* const* d_in, const int* in_sizes, int n_in,
                              void* d_out, int out_size, void* d_ws, size_t ws_size,
                              hipStream_t stream) {
    const int*   tokens = (const int*)  d_in[0];
    const float* tar    = (const float*)d_in[1];
    const float* gen    = (const float*)d_in[2];
    const float* enc    = (const float*)d_in[3];
    const float* dec    = (const float*)d_in[4];
    const float* heads  = (const float*)d_in[5];
    const float* W      = (const float*)d_in[6];
    const float* bias   = (const float*)d_in[7];

    float* fin = (float*)d_out;          // (B,T,V)
    float* ptr = fin + BTV;              // (B,T,V)
    float* pg  = fin + 2 * BTV;          // (B,T)

    float* attn = (float*)d_ws;                  // B*T*I floats (4 MB)
    float* ctx  = attn + (size_t)B * T * I;      // B*T*D floats (4 MB)

    // 1) head mean: B*T*I/4 = 262144 threads
    mean_heads_kernel<<<(B * TI / 4) / 256, 256, 0, stream>>>(heads, attn);

    // 2) WMMA GEMM: 4096 tiles, 4 waves (128 threads) per block -> 1024 blocks
    context_wmma_kernel<<<1024, 128, 0, stream>>>(attn, enc, ctx);

    // 3) p_gen: B*T waves = 2048 -> 65536 threads
    pgen_kernel<<<(B * T * 32) / 256, 256, 0, stream>>>(ctx, dec, tar, W, bias, pg);

    // 4) fused pointer scatter + blend: one WG per (b,t), 128,000 B dynamic LDS
    pointer_fuse_kernel<<<B * T, 512, V * sizeof(float), stream>>>(
        attn, tokens, gen, pg, ptr, fin);
}